// DEEncoder_56264071577828
// MI455X (gfx1250) — compile-verified
//
#include <hip/hip_runtime.h>
#include <hip/hip_bf16.h>

// DynamicEdgeConv encoder for MI455X (gfx1250, wave32).
// All GEMMs run on V_WMMA_F32_16X16X4_F32 (exact fp32): total work is only
// ~28 GFLOP and kNN ordering + final outputs want fp32 precision, so no
// precision downgrade is warranted. Distance matrix is never materialized.
// kNN column slabs are staged once per block into LDS with the gfx1250
// async global->LDS path (ASYNCcnt) since all 4 waves of a block consume
// the identical B-operand tiles.

#define BATCH 16
#define NPTS  2048
#define NPALL (BATCH * NPTS)   // 32768
#define KNN   6

typedef __attribute__((ext_vector_type(2))) float v2f;
typedef __attribute__((ext_vector_type(8))) float v8f;
typedef __attribute__((ext_vector_type(4))) int   v4i;

#define GLOBAL_AS __attribute__((address_space(1)))
#define LDS_AS    __attribute__((address_space(3)))

#if __has_builtin(__builtin_amdgcn_global_load_async_to_lds_b128) && \
    __has_builtin(__builtin_amdgcn_s_wait_asynccnt)
#define HAVE_ASYNC_LDS 1
#else
#define HAVE_ASYNC_LDS 0
#endif

static __device__ __forceinline__ v8f wmma_f32(v2f a, v2f b, v8f c) {
  // (neg_a, A, neg_b, B, c_mod, C, reuse_a, reuse_b)
  return __builtin_amdgcn_wmma_f32_16x16x4_f32(false, a, false, b, (short)0, c,
                                               false, false);
}

static __device__ __forceinline__ v2f mk2(float a, float b) {
  v2f r; r.x = a; r.y = b; return r;
}

// Insert (nd, ni) into ascending top-6 arrays (register resident when unrolled).
static __device__ __forceinline__ void topk_insert(float (&d)[KNN], int (&i)[KNN],
                                                   float nd, int ni) {
  if (nd < d[KNN - 1]) {
    d[KNN - 1] = nd; i[KNN - 1] = ni;
#pragma unroll
    for (int t = KNN - 1; t > 0; --t) {
      if (d[t] < d[t - 1]) {
        float td = d[t]; d[t] = d[t - 1]; d[t - 1] = td;
        int   ti = i[t]; i[t] = i[t - 1]; i[t - 1] = ti;
      }
    }
  }
}

// ---------------------------------------------------------------- row norms
template <int C>
__global__ void rownorm_kernel(const float* __restrict__ x, float* __restrict__ sq) {
  int r = blockIdx.x * blockDim.x + threadIdx.x;
  if (r >= NPALL) return;
  float s = 0.f;
#pragma unroll
  for (int c = 0; c < C; ++c) {
    float v = x[r * C + c];
    s += v * v;
  }
  sq[r] = s;
}

// ----------------------------------------------------------------- fused kNN
// One wave = one 16-row strip of one batch's Gram matrix. 4 waves / block.
// All 4 waves share one batch, so the 16xC column slab is staged once in LDS.
template <int C>
__global__ void knn_kernel(const float* __restrict__ x, const float* __restrict__ sq,
                           int* __restrict__ knn) {
  __shared__ float s_d[4][16][16 * KNN];
  __shared__ int   s_i[4][16][16 * KNN];
  __shared__ __align__(16) float s_xt[16 * 64];   // one 16 x C column slab

  const int tid  = threadIdx.x;
  const int wid  = tid >> 5;
  const int lane = tid & 31;
  const int half = lane >> 4;   // 0: K=0,1 / rows 0-7 of C;  1: K=2,3 / rows 8-15
  const int l16  = lane & 15;

  const int wg = blockIdx.x * 4 + wid;       // 2048 waves total
  const int b  = wg >> 7;                    // batch index (128 row-tiles each)
  const int rt = wg & 127;                   // row tile within batch

  const float* xb = x + (size_t)b * NPTS * C;

  // A operand cache: rows rt*16 + (lane&15), K packed per 16x16x4 layout.
  v2f A[C / 4];
#pragma unroll
  for (int kc = 0; kc < C / 4; ++kc) {
    int k0 = kc * 4 + half * 2;
    const float* p = xb + (rt * 16 + l16) * C + k0;
    A[kc] = mk2(p[0], p[1]);
  }

  float best[8][KNN];
  int   bidx[8][KNN];
#pragma unroll
  for (int v = 0; v < 8; ++v)
#pragma unroll
    for (int t = 0; t < KNN; ++t) { best[v][t] = 3.0e38f; bidx[v][t] = 0; }

  constexpr int TILE_VECS = (16 * C) / 4;    // float4 chunks in a slab

  for (int jt = 0; jt < NPTS / 16; ++jt) {
    const int col = jt * 16 + l16;
    const float sqj = sq[b * NPTS + col];
    if (jt + 1 < NPTS / 16)
      __builtin_prefetch(xb + (col + 16) * C, 0, 1);   // global_prefetch_b8

    // Stage the contiguous 16xC column slab (cols jt*16..+15) into LDS once.
    const float* gsrc = xb + (size_t)(jt * 16) * C;
#if HAVE_ASYNC_LDS
    for (int i = tid; i < TILE_VECS; i += 128)
      __builtin_amdgcn_global_load_async_to_lds_b128(
          (GLOBAL_AS v4i*)(gsrc + i * 4), (LDS_AS v4i*)(s_xt + i * 4), 0, 0);
    __builtin_amdgcn_s_wait_asynccnt(0);
#else
    for (int i = tid; i < TILE_VECS; i += 128) {
      float4 vdat = *(const float4*)(gsrc + i * 4);
      *(float4*)(s_xt + i * 4) = vdat;
    }
#endif
    __syncthreads();   // slab visible to all waves

    // B operands from LDS (ds_load_b64 per K-step).
    v2f Bm[C / 4];
#pragma unroll
    for (int kc = 0; kc < C / 4; ++kc) {
      int k0 = kc * 4 + half * 2;
      const float* p = &s_xt[l16 * C + k0];
      Bm[kc] = mk2(p[0], p[1]);
    }

    v8f acc = {};
#pragma unroll
    for (int kc = 0; kc < C / 4; ++kc) acc = wmma_f32(A[kc], Bm[kc], acc);

    // score_j = |x_j|^2 - 2 x_i.x_j   (|x_i|^2 constant per row: dropped)
#pragma unroll
    for (int v = 0; v < 8; ++v) {
      int rowg = rt * 16 + v + half * 8;
      float score = sqj - 2.0f * acc[v];
      if (rowg != col) topk_insert(best[v], bidx[v], score, col);
    }
    __syncthreads();   // everyone done with the slab before next overwrite
  }

  // Per-wave merge: 16 lanes x top-6 candidates per row -> global top-6.
#pragma unroll
  for (int v = 0; v < 8; ++v)
#pragma unroll
    for (int t = 0; t < KNN; ++t) {
      s_d[wid][v + half * 8][l16 * KNN + t] = best[v][t];
      s_i[wid][v + half * 8][l16 * KNN + t] = bidx[v][t];
    }
  __builtin_amdgcn_wave_barrier();   // same-wave LDS ops are in-order

  if (lane < 16) {
    float bd[KNN]; int bi[KNN];
#pragma unroll
    for (int t = 0; t < KNN; ++t) { bd[t] = 3.0e38f; bi[t] = 0; }
    for (int c = 0; c < 16 * KNN; ++c)
      topk_insert(bd, bi, s_d[wid][lane][c], s_i[wid][lane][c]);
    int row = b * NPTS + rt * 16 + lane;
#pragma unroll
    for (int t = 0; t < KNN; ++t) knn[row * KNN + t] = bi[t];
  }
}

// ------------------------------------------------------------- edge MLP + max
// K=6 padded to 8 edges/point  ->  one 16-row M-tile = 2 points; neighbor max
// collapses to a per-lane max over the 8 accumulator VGPRs.
template <int CIN, int HH, bool RES>
__global__ void edge_mlp_kernel(const float* __restrict__ x, const int* __restrict__ knn,
                                const float* __restrict__ W1, const float* __restrict__ b1,
                                const float* __restrict__ W2, const float* __restrict__ b2,
                                const float* __restrict__ res, float* __restrict__ out) {
  constexpr int K2 = 2 * CIN;
  __shared__ float s_h1[8][16][64];

  const int tid  = threadIdx.x;
  const int wid  = tid >> 5;
  const int lane = tid & 31;
  const int half = lane >> 4;
  const int l16  = lane & 15;

  const int wg = blockIdx.x * 8 + wid;   // 16384 waves, 2 points each
  const int p0 = wg * 2;

  // This lane's A-operand row: m = l16 -> point pm, edge slot s (6,7 dup 5).
  const int m  = l16;
  const int pm = p0 + (m >> 3);
  int s = m & 7; if (s > 5) s = 5;
  const int bbase = pm & ~(NPTS - 1);
  const int j = knn[pm * KNN + s];

  const float* xp = x + (size_t)pm * CIN;
  const float* xj = x + (size_t)(bbase + j) * CIN;

  // msg = [x_i , x_j - x_i]  packed into A layout for 16x16x4.
  v2f A1[K2 / 4];
#pragma unroll
  for (int kc = 0; kc < K2 / 4; ++kc) {
    int k0 = kc * 4 + half * 2;
    float a0 = (k0 < CIN) ? xp[k0] : (xj[k0 - CIN] - xp[k0 - CIN]);
    int k1 = k0 + 1;
    float a1 = (k1 < CIN) ? xp[k1] : (xj[k1 - CIN] - xp[k1 - CIN]);
    A1[kc] = mk2(a0, a1);
  }

  // Phase 1: h1 = relu(msg @ W1 + b1)  (M=16, K=2*CIN, N=HH)
#pragma unroll
  for (int nt = 0; nt < HH / 16; ++nt) {
    int n = nt * 16 + l16;
    v8f acc = {};
#pragma unroll
    for (int kc = 0; kc < K2 / 4; ++kc) {
      int k0 = kc * 4 + half * 2;
      v2f bw = mk2(W1[k0 * HH + n], W1[(k0 + 1) * HH + n]);
      acc = wmma_f32(A1[kc], bw, acc);
    }
    float bias = b1[n];
#pragma unroll
    for (int v = 0; v < 8; ++v) {
      float h = acc[v] + bias;
      s_h1[wid][v + half * 8][n] = h > 0.f ? h : 0.f;
    }
  }
  __builtin_amdgcn_wave_barrier();

  // Phase 2: h2 = h1 @ W2, then per-point max over 8 padded edges.
  v2f A2[HH / 4];
#pragma unroll
  for (int kc = 0; kc < HH / 4; ++kc) {
    int k0 = kc * 4 + half * 2;
    A2[kc] = mk2(s_h1[wid][m][k0], s_h1[wid][m][k0 + 1]);
  }

  const int pout = p0 + half;   // lanes 0-15: rows 0-7 (pt A); 16-31: rows 8-15 (pt B)
#pragma unroll
  for (int nt = 0; nt < HH / 16; ++nt) {
    int n = nt * 16 + l16;
    v8f acc = {};
#pragma unroll
    for (int kc = 0; kc < HH / 4; ++kc) {
      int k0 = kc * 4 + half * 2;
      v2f bw = mk2(W2[k0 * HH + n], W2[(k0 + 1) * HH + n]);
      acc = wmma_f32(A2[kc], bw, acc);
    }
    float mv = acc[0];
#pragma unroll
    for (int v = 1; v < 8; ++v) mv = fmaxf(mv, acc[v]);
    float val = mv + b2[n];
    if (RES) val += res[(size_t)pout * HH + n];
    out[(size_t)pout * HH + n] = fmaxf(val, 0.f);
  }
}

extern "C" void kernel_launch(void* const* d_in, const int* in_sizes, int n_in,
                              void* d_out, int out_size, void* d_ws, size_t ws_size,
                              hipStream_t stream) {
  (void)in_sizes; (void)n_in; (void)out_size; (void)ws_size;
  const float* x    = (const float*)d_in[0];
  const float* W1_0 = (const float*)d_in[2];
  const float* b1_0 = (const float*)d_in[3];
  const float* W2_0 = (const float*)d_in[4];
  const float* b2_0 = (const float*)d_in[5];
  const float* W1_1 = (const float*)d_in[6];
  const float* b1_1 = (const float*)d_in[7];
  const float* W2_1 = (const float*)d_in[8];
  const float* b2_1 = (const float*)d_in[9];
  const float* W1_2 = (const float*)d_in[10];
  const float* b1_2 = (const float*)d_in[11];
  const float* W2_2 = (const float*)d_in[12];
  const float* b2_2 = (const float*)d_in[13];

  char* ws = (char*)d_ws;
  float* x0 = (float*)(ws);                                // 32768*64*4 = 8 MB
  float* x1 = (float*)(ws + (size_t)8 * 1024 * 1024);      // 32768*32*4 = 4 MB
  float* sq = (float*)(ws + (size_t)12 * 1024 * 1024);     // 128 KB
  int*   kn = (int*)  (ws + (size_t)13 * 1024 * 1024);     // 768 KB
  float* out = (float*)d_out;

  dim3 bNorm(256), gNorm(NPALL / 256);
  dim3 bKnn(128),  gKnn(2048 / 4);     // 2048 waves, 4 per block
  dim3 bMlp(256),  gMlp(NPALL / 16);   // 16 points per block (8 waves x 2)

  // Layer 0: C=32 -> 64
  rownorm_kernel<32><<<gNorm, bNorm, 0, stream>>>(x, sq);
  knn_kernel<32><<<gKnn, bKnn, 0, stream>>>(x, sq, kn);
  edge_mlp_kernel<32, 64, false><<<gMlp, bMlp, 0, stream>>>(
      x, kn, W1_0, b1_0, W2_0, b2_0, nullptr, x0);

  // Layer 1: C=64 -> 32
  rownorm_kernel<64><<<gNorm, bNorm, 0, stream>>>(x0, sq);
  knn_kernel<64><<<gKnn, bKnn, 0, stream>>>(x0, sq, kn);
  edge_mlp_kernel<64, 32, false><<<gMlp, bMlp, 0, stream>>>(
      x0, kn, W1_1, b1_1, W2_1, b2_1, nullptr, x1);

  // Layer 2: C=32 -> 64, + x0 residual, relu
  rownorm_kernel<32><<<gNorm, bNorm, 0, stream>>>(x1, sq);
  knn_kernel<32><<<gKnn, bKnn, 0, stream>>>(x1, sq, kn);
  edge_mlp_kernel<32, 64, true><<<gMlp, bMlp, 0, stream>>>(
      x1, kn, W1_2, b1_2, W2_2, b2_2, x0, out);
}